// FTT_Layer_34978213659277
// MI455X (gfx1250) — compile-verified
//
#include <hip/hip_runtime.h>

#define S_TOTAL 65536
#define RANK 32

typedef __attribute__((ext_vector_type(2))) float v2f;
typedef __attribute__((ext_vector_type(8))) float v8f;

// ---------------- Phase 0: zero workspace accumulators ----------------
__global__ void zero_ws_kernel(float* __restrict__ ws, int n) {
  int i = blockIdx.x * blockDim.x + threadIdx.x;
  if (i < n) ws[i] = 0.0f;
}

// ---------------- Phase A: extra[d,e] = sum_s z[s]*core[d,s,e] ----------------
// core layout (32, S, 32). One block per (s-chunk, d). 256 threads = 8 waves.
// Per wave-iteration: 4 consecutive s-rows; lane -> (row = lane>>3, e = 4*(lane&7)),
// i.e. one fully-coalesced 512B b128 load per wave. Streaming, HBM-bound.
__global__ void reduce_core_kernel(const float* __restrict__ z,
                                   const float* __restrict__ core,
                                   float* __restrict__ extra,   // 32x32
                                   int s_per_block) {
  const int d    = blockIdx.y;
  const int s0   = blockIdx.x * s_per_block;
  const int lane = threadIdx.x & 31;
  const int wave = threadIdx.x >> 5;
  const int grp  = lane >> 3;             // row within the 4-row tile
  const int ecol = (lane & 7) * 4;        // e base (float4)

  const float* __restrict__ base = core + (size_t)d * S_TOTAL * RANK;

  float acc0 = 0.f, acc1 = 0.f, acc2 = 0.f, acc3 = 0.f;
  const int send = s0 + s_per_block;
  for (int s = s0 + wave * 4; s < send; s += 32) {   // 8 waves * 4 rows
    const int srow = s + grp;
    __builtin_prefetch(base + (size_t)(srow + 32) * RANK + ecol, 0, 3);
    const float4 val = *(const float4*)(base + (size_t)srow * RANK + ecol);
    const float  zz  = z[srow];
    acc0 = fmaf(zz, val.x, acc0);
    acc1 = fmaf(zz, val.y, acc1);
    acc2 = fmaf(zz, val.z, acc2);
    acc3 = fmaf(zz, val.w, acc3);
  }

  __shared__ float red[RANK];
  if (threadIdx.x < RANK) red[threadIdx.x] = 0.f;
  __syncthreads();
  atomicAdd(&red[ecol + 0], acc0);   // ds_add_f32
  atomicAdd(&red[ecol + 1], acc1);
  atomicAdd(&red[ecol + 2], acc2);
  atomicAdd(&red[ecol + 3], acc3);
  __syncthreads();
  if (threadIdx.x < RANK)
    atomicAdd(&extra[d * RANK + threadIdx.x], red[threadIdx.x]);
}

// ---------------- Phase A': extra3[d] = sum_s z[s]*core3[d,s,0] ----------------
__global__ void reduce_core3_kernel(const float* __restrict__ z,
                                    const float* __restrict__ core3,
                                    float* __restrict__ extra3,  // 32
                                    int s_per_block) {
  const int d  = blockIdx.y;
  const int s0 = blockIdx.x * s_per_block;
  const float* __restrict__ base = core3 + (size_t)d * S_TOTAL;

  float acc = 0.f;
  const int send = s0 + s_per_block;
  for (int s = s0 + threadIdx.x * 4; s < send; s += blockDim.x * 4) {
    const float4 c = *(const float4*)(base + s);
    const float4 zz = *(const float4*)(z + s);
    acc = fmaf(zz.x, c.x, acc);
    acc = fmaf(zz.y, c.y, acc);
    acc = fmaf(zz.z, c.z, acc);
    acc = fmaf(zz.w, c.w, acc);
  }
  __shared__ float red[256];
  red[threadIdx.x] = acc;
  __syncthreads();
  for (int off = 128; off > 0; off >>= 1) {
    if (threadIdx.x < off) red[threadIdx.x] += red[threadIdx.x + off];
    __syncthreads();
  }
  if (threadIdx.x == 0) atomicAdd(&extra3[d], red[0]);
}

// ---------------- Phase B: v = (E1 @ E2) @ extra3 via V_WMMA_F32_16X16X4_F32 ----
// 128 threads = 4 waves; wave w owns the 16x16 tile (bi,bj) of M = E1@E2.
// K=32 stepped 4 at a time: 8 accumulating WMMAs per wave, all f32.
__global__ void combine_kernel(const float* __restrict__ extra1,
                               const float* __restrict__ extra2,
                               const float* __restrict__ extra3,
                               float* __restrict__ v_out) {
  __shared__ float E1[RANK * RANK];
  __shared__ float E2[RANK * RANK];
  __shared__ float M[RANK * RANK];
  for (int i = threadIdx.x; i < RANK * RANK; i += 128) {
    E1[i] = extra1[i];
    E2[i] = extra2[i];
  }
  __syncthreads();

  const int wave  = threadIdx.x >> 5;
  const int lane  = threadIdx.x & 31;
  const int bi    = (wave >> 1) * 16;     // tile row of M
  const int bj    = (wave & 1) * 16;      // tile col of M
  const int row   = lane & 15;
  const int khalf = (lane >> 4) * 2;      // ISA 16x4 f32 A layout: hi lanes hold K=2,3

  v8f c = {0.f, 0.f, 0.f, 0.f, 0.f, 0.f, 0.f, 0.f};
  for (int kc = 0; kc < RANK; kc += 4) {
    v2f a, b;
    // A 16x4 (MxK): lane = row, VGPR i -> K = khalf + i
    a.x = E1[(bi + row) * RANK + kc + khalf + 0];
    a.y = E1[(bi + row) * RANK + kc + khalf + 1];
    // B 4x16 (KxN): lane = col, VGPR i -> K = khalf + i
    b.x = E2[(kc + khalf + 0) * RANK + bj + row];
    b.y = E2[(kc + khalf + 1) * RANK + bj + row];
    // (neg_a, A, neg_b, B, c_mod, C, reuse_a, reuse_b)
    c = __builtin_amdgcn_wmma_f32_16x16x4_f32(false, a, false, b,
                                              (short)0, c, false, false);
  }

  // C/D 16x16 f32 layout: VGPR r -> row r (lanes 0-15) / row r+8 (lanes 16-31)
  {
    const int mrow0 = (lane >> 4) * 8;
    const int ncol  = lane & 15;
    for (int r = 0; r < 8; ++r)
      M[(bi + mrow0 + r) * RANK + (bj + ncol)] = c[r];
  }
  __syncthreads();

  if (threadIdx.x < RANK) {
    float acc = 0.f;
    for (int e = 0; e < RANK; ++e)
      acc = fmaf(M[threadIdx.x * RANK + e], extra3[e], acc);
    v_out[threadIdx.x] = acc;
  }
}

// ---------------- Phase C: out[s] = dot(core0[s,:], v) ----------------
// One wave per output row: lane e reads core0[s*32+e] (coalesced 128B),
// wave32 shuffle reduction.
__global__ void final_kernel(const float* __restrict__ core0,
                             const float* __restrict__ v,
                             float* __restrict__ out) {
  const int gtid = blockIdx.x * blockDim.x + threadIdx.x;
  const int s    = gtid >> 5;
  const int lane = threadIdx.x & 31;
  float p = core0[(size_t)s * RANK + lane] * v[lane];
  for (int off = 16; off > 0; off >>= 1)
    p += __shfl_xor(p, off, 32);
  if (lane == 0) out[s] = p;
}

extern "C" void kernel_launch(void* const* d_in, const int* in_sizes, int n_in,
                              void* d_out, int out_size, void* d_ws, size_t ws_size,
                              hipStream_t stream) {
  const float* z     = (const float*)d_in[0];
  const float* core0 = (const float*)d_in[1];
  const float* core1 = (const float*)d_in[2];
  const float* core2 = (const float*)d_in[3];
  const float* core3 = (const float*)d_in[4];
  float* out = (float*)d_out;
  float* ws  = (float*)d_ws;

  float* extra1 = ws;                    // 1024 floats
  float* extra2 = ws + 1024;             // 1024 floats
  float* extra3 = ws + 2048;             // 32 floats
  float* vvec   = ws + 2080;             // 32 floats
  const int n_zero = 2112;

  zero_ws_kernel<<<(n_zero + 255) / 256, 256, 0, stream>>>(ws, n_zero);

  const int S_CHUNKS = 32;               // 32 chunks x 32 d = 1024 blocks / core
  const int spb = S_TOTAL / S_CHUNKS;    // 2048 rows per block
  dim3 gridA(S_CHUNKS, RANK);
  reduce_core_kernel<<<gridA, 256, 0, stream>>>(z, core1, extra1, spb);
  reduce_core_kernel<<<gridA, 256, 0, stream>>>(z, core2, extra2, spb);

  dim3 gridA3(4, RANK);
  reduce_core3_kernel<<<gridA3, 256, 0, stream>>>(z, core3, extra3, S_TOTAL / 4);

  combine_kernel<<<1, 128, 0, stream>>>(extra1, extra2, extra3, vvec);

  final_kernel<<<S_TOTAL / 8, 256, 0, stream>>>(core0, vvec, out);
}